// RNNModel_32126355374416
// MI455X (gfx1250) — compile-verified
//
#include <hip/hip_runtime.h>

typedef __attribute__((ext_vector_type(2))) float v2f;
typedef __attribute__((ext_vector_type(8))) float v8f;

#define RNN_B 4096
#define RNN_T 512
#define RNN_I 16
#define RNN_H 6

__device__ __forceinline__ v8f wmma_f32_4(v2f a, v2f b, v8f c) {
  // D(16x16,f32) = A(16x4,f32) * B(4x16,f32) + C
  return __builtin_amdgcn_wmma_f32_16x16x4_f32(false, a, false, b, (short)0, c,
                                               false, false);
}

__device__ __forceinline__ float tanh_fast(float x) {
#if __has_builtin(__builtin_amdgcn_tanhf)
  return __builtin_amdgcn_tanhf(x);  // v_tanh_f32: saturates for large |x|
#else
  x = __builtin_amdgcn_fmed3f(x, -9.0f, 9.0f);  // single-instr clamp for exp path
  float e = __expf(2.0f * x);
  return (e - 1.0f) * __builtin_amdgcn_rcpf(e + 1.0f);
#endif
}

__device__ __forceinline__ void tanh8(v8f& a) {
#pragma unroll
  for (int i = 0; i < 8; ++i) a[i] = tanh_fast(a[i]);
}

__global__ __launch_bounds__(64) void rnn_fused_wmma(
    const float* __restrict__ x, const float* __restrict__ W_ih,
    const float* __restrict__ W_hh, const float* __restrict__ b_ih,
    const float* __restrict__ b_hh, const float* __restrict__ fc_w,
    const float* __restrict__ fc_b, float* __restrict__ out) {
  // One wave32 owns 16 batch rows for all 512 steps.
  __shared__ __align__(16) float hbuf[2][16][18];  // stride 18: 8B-aligned b64, bank-clean

  const int tid  = threadIdx.x;
  const int w    = tid >> 5;
  const int lane = tid & 31;
  const int n    = lane & 15;  // column (hidden) for B/C/D; row (batch) for A
  const int hi   = lane >> 4;  // half-wave selector
  const int b0   = (blockIdx.x * 2 + w) * 16;

  // ---- Resident B-operands (zero-padded to N=16 / K=8) ----
  v2f bih[4];
#pragma unroll
  for (int s = 0; s < 4; ++s) {
#pragma unroll
    for (int v = 0; v < 2; ++v) {
      int k = 4 * s + 2 * hi + v;  // B VGPR v holds row K = 4s + 2*hi + v
      bih[s][v] = (n < RNN_H) ? W_ih[n * RNN_I + k] : 0.0f;  // W_ih^T[k][n]
    }
  }
  v2f whh[2];
#pragma unroll
  for (int s = 0; s < 2; ++s) {
#pragma unroll
    for (int v = 0; v < 2; ++v) {
      int k = 4 * s + 2 * hi + v;
      whh[s][v] = (n < RNN_H && k < RNN_H) ? W_hh[n * RNN_H + k] : 0.0f;
    }
  }
  const float bn = (n < RNN_H) ? (b_ih[n] + b_hh[n]) : 0.0f;
  v8f cbias;
#pragma unroll
  for (int i = 0; i < 8; ++i) cbias[i] = bn;

  // A-operand x pointer: lane holds rows m = lane&15, K-pair starting at 2*hi
  const float* xp = x + (size_t)(b0 + n) * (RNN_T * RNN_I) + 2 * hi;

  v2f ha0 = 0.0f, ha1 = 0.0f;  // h_0 = 0 (A-layout slices, K 0..3 / 4..7)
  v2f xa[4], xb[4];
#pragma unroll
  for (int s = 0; s < 4; ++s) xa[s] = *(const v2f*)(xp + 4 * s);

  float(*hb)[18] = hbuf[w];

  for (int t = 0; t < RNN_T; t += 2) {
    // ---------------- step t (operands in xa) ----------------
    v8f acc = cbias;
#pragma unroll
    for (int s = 0; s < 4; ++s) acc = wmma_f32_4(xa[s], bih[s], acc);
    acc = wmma_f32_4(ha0, whh[0], acc);
    acc = wmma_f32_4(ha1, whh[1], acc);
    {  // prefetch x for t+1 before the LDS round-trip
      const float* p = xp + (size_t)(t + 1) * RNN_I;
#pragma unroll
      for (int s = 0; s < 4; ++s) xb[s] = *(const v2f*)(p + 4 * s);
    }
    tanh8(acc);
    // D-layout -> LDS (row r+8*hi, col n), then reload in A-layout
#pragma unroll
    for (int r = 0; r < 8; ++r) hb[r + hi * 8][n] = acc[r];
    __builtin_amdgcn_wave_barrier();
    ha0 = *(const v2f*)&hb[n][2 * hi];
    ha1 = *(const v2f*)&hb[n][4 + 2 * hi];
    __builtin_amdgcn_wave_barrier();

    // ---------------- step t+1 (operands in xb) ----------------
    acc = cbias;
#pragma unroll
    for (int s = 0; s < 4; ++s) acc = wmma_f32_4(xb[s], bih[s], acc);
    acc = wmma_f32_4(ha0, whh[0], acc);
    acc = wmma_f32_4(ha1, whh[1], acc);
    {
      const int t2 = (t + 2) & (RNN_T - 1);  // power-of-two wrap: harmless dup load
      const float* p = xp + (size_t)t2 * RNN_I;
#pragma unroll
      for (int s = 0; s < 4; ++s) xa[s] = *(const v2f*)(p + 4 * s);
    }
    tanh8(acc);
#pragma unroll
    for (int r = 0; r < 8; ++r) hb[r + hi * 8][n] = acc[r];
    __builtin_amdgcn_wave_barrier();
    ha0 = *(const v2f*)&hb[n][2 * hi];
    ha1 = *(const v2f*)&hb[n][4 + 2 * hi];
    __builtin_amdgcn_wave_barrier();
  }

  // ---- FC head: out[b] = h_last . fc_w + fc_b (h_last sits batch-major in LDS) ----
  if (lane < 16) {
    float s = fc_b[0];
#pragma unroll
    for (int j = 0; j < RNN_H; ++j) s += hb[lane][j] * fc_w[j];
    out[b0 + lane] = s;
  }
}

extern "C" void kernel_launch(void* const* d_in, const int* in_sizes, int n_in,
                              void* d_out, int out_size, void* d_ws,
                              size_t ws_size, hipStream_t stream) {
  (void)in_sizes; (void)n_in; (void)out_size; (void)d_ws; (void)ws_size;
  const float* x    = (const float*)d_in[0];
  const float* W_ih = (const float*)d_in[1];
  const float* W_hh = (const float*)d_in[2];
  const float* b_ih = (const float*)d_in[3];
  const float* b_hh = (const float*)d_in[4];
  const float* fc_w = (const float*)d_in[5];
  const float* fc_b = (const float*)d_in[6];
  float* out = (float*)d_out;

  dim3 block(64);                 // 2 waves, each owns 16 batches + private LDS slice
  dim3 grid(RNN_B / 32);          // 128 blocks -> spread across WGPs
  rnn_fused_wmma<<<grid, block, 0, stream>>>(x, W_ih, W_hh, b_ih, b_hh, fc_w,
                                             fc_b, out);
}